// SMPL_layer_64536178590095
// MI455X (gfx1250) — compile-verified
//
#include <hip/hip_runtime.h>
#include <hip/hip_bf16.h>

typedef float v2f __attribute__((ext_vector_type(2)));
typedef float v4f __attribute__((ext_vector_type(4)));
typedef float v8f __attribute__((ext_vector_type(8)));

#define SMPL_V   6890
#define SMPL_N3  20670   // V*3
#define VPS      20672   // padded vposed row stride (16B-aligned rows)
#define SMPL_J   24
#define SMPL_NB  10
#define KPAD     220     // 11 betas+kid, [11]=0, 12..218 pose feature, [219]=0

__device__ __forceinline__ int imin(int a, int b) { return a < b ? a : b; }

// ---------------------------------------------------------------------------
// Kernel A: JT[j,k] = sum_v Jr[j,v]*vt[v,k];  JS[j,k,l] = sum_v Jr[j,v]*sd[v,k,l]
// grid = 72 blocks (j*3+k), 256 threads
// ---------------------------------------------------------------------------
__global__ __launch_bounds__(256) void jreg_kernel(const float* __restrict__ Jr,
                                                   const float* __restrict__ vt,
                                                   const float* __restrict__ sd,
                                                   float* __restrict__ JTJS) {
    int jk = blockIdx.x;
    int j = jk / 3, k = jk % 3;
    int tid = threadIdx.x;
    float acc[12];
#pragma unroll
    for (int i = 0; i < 12; ++i) acc[i] = 0.f;
    for (int v = tid; v < SMPL_V; v += 256) {
        float w = Jr[(size_t)j * SMPL_V + v];
        acc[0] += w * vt[v * 3 + k];
        const float* s = sd + (size_t)(v * 3 + k) * 11;
#pragma unroll
        for (int l = 0; l < 11; ++l) acc[1 + l] += w * s[l];
    }
    __shared__ float red[256];
    for (int i = 0; i < 12; ++i) {
        red[tid] = acc[i];
        __syncthreads();
        for (int s = 128; s > 0; s >>= 1) {
            if (tid < s) red[tid] += red[tid + s];
            __syncthreads();
        }
        if (tid == 0) {
            if (i == 0) JTJS[jk] = red[0];
            else        JTJS[72 + jk * 11 + (i - 1)] = red[0];
        }
        __syncthreads();
    }
}

// ---------------------------------------------------------------------------
// Kernel PF: padded pose-feature matrix PF[b, 0..219]
//   0..9 betas, 10 kid, 11 zero, 12..218 (poses[:,1:]-I).flat, 219 zero
// ---------------------------------------------------------------------------
__global__ __launch_bounds__(256) void pf_kernel(const float* __restrict__ poses,
                                                 const float* __restrict__ betas,
                                                 const float* __restrict__ kidb,
                                                 float* __restrict__ PF) {
    int b = blockIdx.x;
    int t = threadIdx.x;
    if (t < KPAD) {
        float v;
        if (t < 10)       v = betas[(size_t)b * 10 + t];
        else if (t == 10) v = kidb[b];
        else if (t >= 12 && t < 219) {
            int q = t - 12;                       // 0..206
            float p = poses[(size_t)b * 216 + 9 + q];
            int e = q % 9;
            v = p - ((e == 0 || e == 4 || e == 8) ? 1.0f : 0.0f);
        } else v = 0.f;                           // t==11 or t==219
        PF[(size_t)b * KPAD + t] = v;
    }
}

// ---------------------------------------------------------------------------
// Kernel B: per-batch joints, kinematic chain, A matrices, root, poses copy.
// grid = B blocks, 32 threads (one wave)
// ---------------------------------------------------------------------------
__global__ __launch_bounds__(32) void joints_kernel(const float* __restrict__ poses,
                                                    const float* __restrict__ betas,
                                                    const float* __restrict__ kidb,
                                                    const int*   __restrict__ parents,
                                                    const float* __restrict__ JTJS,
                                                    float* __restrict__ joints_out,
                                                    float* __restrict__ poses_out,
                                                    float* __restrict__ Amat,
                                                    float* __restrict__ rootp) {
    int b = blockIdx.x;
    int t = threadIdx.x;
    __shared__ float fb[11];
    __shared__ float Jp[SMPL_J][3];
    __shared__ float Tl[SMPL_J][16];
    __shared__ float Tg[SMPL_J][16];

    if (t < 11) fb[t] = (t < 10) ? betas[(size_t)b * 10 + t] : kidb[b];
    __syncthreads();

    for (int i = t; i < 72; i += 32) {
        float a = JTJS[i];
        const float* js = JTJS + 72 + i * 11;
#pragma unroll
        for (int l = 0; l < 11; ++l) a += fb[l] * js[l];
        Jp[i / 3][i % 3] = a;
    }
    __syncthreads();

    for (int i = t; i < SMPL_J * 16; i += 32) {
        int j = i >> 4, mn = i & 15, m = mn >> 2, n = mn & 3;
        float val;
        if (m < 3 && n < 3) {
            val = poses[(size_t)b * 216 + j * 9 + m * 3 + n];
        } else if (m < 3) {
            int p = parents[j];
            val = (j == 0) ? Jp[0][m] : (Jp[j][m] - Jp[p][m]);
        } else {
            val = (n == 3) ? 1.f : 0.f;
        }
        Tl[j][mn] = val;
    }
    __syncthreads();

    if (t < 16) Tg[0][t] = Tl[0][t];
    __syncthreads();
    for (int j = 1; j < SMPL_J; ++j) {
        int p = parents[j];
        if (t < 16) {
            int m = t >> 2, n = t & 3;
            float s = 0.f;
#pragma unroll
            for (int kk = 0; kk < 4; ++kk) s += Tg[p][m * 4 + kk] * Tl[j][kk * 4 + n];
            Tg[j][t] = s;
        }
        __syncthreads();
    }

    if (t < 3) rootp[b * 4 + t] = Tg[0][t * 4 + 3];

    for (int i = t; i < 72; i += 32) {
        int j = i / 3, k = i % 3;
        joints_out[(size_t)b * 72 + i] = Tg[j][k * 4 + 3] - Tg[0][k * 4 + 3];
    }

    for (int i = t; i < SMPL_J * 12; i += 32) {
        int j = i / 12, mn = i % 12, m = mn >> 2, n = mn & 3;
        float val = Tg[j][m * 4 + n];
        if (n == 3) {
#pragma unroll
            for (int c = 0; c < 3; ++c) val -= Tg[j][m * 4 + c] * Jp[j][c];
        }
        Amat[(size_t)b * 288 + i] = val;
    }

    for (int i = t; i < 216; i += 32)
        poses_out[(size_t)b * 216 + i] = poses[(size_t)b * 216 + i];
}

// ---------------------------------------------------------------------------
// Kernel C: fused shape+pose blend GEMM, transposed D(n, b).
//   vposed[b, n] = vt[n] + sum_k Bmat[k,n] * PF[b,k]
// Divergent indices are unsigned 32-bit (uniform-base + zext offset -> GVS
// saddr addressing); last K-chunk peeled so the steady loop has no clamps.
// grid = (ceil(N3/512), B/16), block = 128; each wave: 128 n x 16 batch.
// ---------------------------------------------------------------------------
__global__ __launch_bounds__(128) void posegemm_kernel(const float* __restrict__ PF,
                                                       const float* __restrict__ pd,
                                                       const float* __restrict__ sd,
                                                       const float* __restrict__ vt,
                                                       float* __restrict__ vposed) {
    int wave  = threadIdx.x >> 5;
    int lane  = threadIdx.x & 31;
    int nbase = blockIdx.x * 512 + wave * 128;
    int bbase = blockIdx.y * 16;
    int l16   = lane & 15;
    int khalf = lane >> 4;

    v8f acc[8];
#pragma unroll
    for (int s = 0; s < 8; ++s) acc[s] = (v8f)(0.f);

    const float* pfrow = PF + (size_t)(bbase + l16) * KPAD;   // B-fragment src

    unsigned ncl[8], sdo[8];                                  // u32 lane offsets
#pragma unroll
    for (int s = 0; s < 8; ++s) {
        ncl[s] = (unsigned)imin(nbase + s * 16 + l16, SMPL_N3 - 1);
        sdo[s] = ncl[s] * 11u;
    }

    // Region 1: shapedirs (K rows 0..11; row 11 don't-care, PF[.,11]==0)
#pragma unroll
    for (int k0 = 0; k0 < 12; k0 += 4) {
        int kr0 = k0 + 2 * khalf;
        v2f bb = *(const v2f*)(pfrow + kr0);
        unsigned l0 = (unsigned)imin(kr0, 10);
        unsigned l1 = (unsigned)imin(kr0 + 1, 10);
#pragma unroll
        for (int s = 0; s < 8; ++s) {
            v2f a;
            a.x = sd[sdo[s] + l0];
            a.y = sd[sdo[s] + l1];
            acc[s] = __builtin_amdgcn_wmma_f32_16x16x4_f32(
                false, a, false, bb, (short)0, acc[s], false, false);
        }
    }

    // Region 2 steady state: posedirs rows 0..203, no clamps needed
    for (int k0 = 12; k0 < 216; k0 += 4) {
        int kr0 = k0 + 2 * khalf;
        v2f bb = *(const v2f*)(pfrow + kr0);
        const float* p0 = pd + (size_t)(kr0 - 12) * SMPL_N3;
        const float* p1 = p0 + SMPL_N3;
#pragma unroll
        for (int s = 0; s < 8; ++s) {
            v2f a;
            a.x = p0[ncl[s]];
            a.y = p1[ncl[s]];
            acc[s] = __builtin_amdgcn_wmma_f32_16x16x4_f32(
                false, a, false, bb, (short)0, acc[s], false, false);
        }
    }

    // Region 2 tail (k0 = 216): rows 204..206 + don't-care row 219 (PF==0)
    {
        int kr0 = 216 + 2 * khalf;                 // 216 or 218
        v2f bb = *(const v2f*)(pfrow + kr0);
        const float* p0 = pd + (size_t)(kr0 - 12) * SMPL_N3;
        const float* p1 = pd + (size_t)(imin(kr0 + 1, 218) - 12) * SMPL_N3;
#pragma unroll
        for (int s = 0; s < 8; ++s) {
            v2f a;
            a.x = p0[ncl[s]];
            a.y = p1[ncl[s]];
            acc[s] = __builtin_amdgcn_wmma_f32_16x16x4_f32(
                false, a, false, bb, (short)0, acc[s], false, false);
        }
    }

    // Store: lane owns batch row (bbase+l16), n = nbase + s*16 + 8*khalf + r
    float* obase = vposed + (size_t)(bbase + l16) * VPS;
#pragma unroll
    for (int s = 0; s < 8; ++s) {
        int n0 = nbase + s * 16 + 8 * khalf;      // 8 consecutive n, 32B aligned
        if (n0 + 7 < SMPL_N3) {
            v4f vtl = *(const v4f*)(vt + n0);
            v4f vth = *(const v4f*)(vt + n0 + 4);
            v4f lo = { acc[s][0] + vtl[0], acc[s][1] + vtl[1],
                       acc[s][2] + vtl[2], acc[s][3] + vtl[3] };
            v4f hi = { acc[s][4] + vth[0], acc[s][5] + vth[1],
                       acc[s][6] + vth[2], acc[s][7] + vth[3] };
            *(v4f*)(obase + n0)     = lo;
            *(v4f*)(obase + n0 + 4) = hi;
        } else {
#pragma unroll
            for (int r = 0; r < 8; ++r) {
                int n = n0 + r;
                if (n < SMPL_N3) obase[n] = acc[s][r] + vt[n];
            }
        }
    }
}

// ---------------------------------------------------------------------------
// Kernel D: LBS.  T = W_{16x24} @ A_b_{24x12pad16} via WMMA, then
//   verts[b,v,:] = R*v_posed + t - root
// grid = (ceil(V/128), B), block = 256 (8 waves, 16 vertices each)
// ---------------------------------------------------------------------------
__global__ __launch_bounds__(256) void lbs_kernel(const float* __restrict__ W,
                                                  const float* __restrict__ Amat,
                                                  const float* __restrict__ vposed,
                                                  const float* __restrict__ rootp,
                                                  float* __restrict__ verts) {
    int b     = blockIdx.y;
    int wave  = threadIdx.x >> 5;
    int lane  = threadIdx.x & 31;
    int vtile = blockIdx.x * 128 + wave * 16;
    int mrow  = lane & 15;
    int khalf = lane >> 4;
    int c     = lane & 15;                 // affine component column (0..11 used)
    unsigned ccl = (unsigned)imin(c, 11);  // clamped (cols 12..15 don't-care)

    __shared__ float T[8][16][16];

    const float* Ab   = Amat + (size_t)b * 288;
    const float* wrow = W + (size_t)imin(vtile + mrow, SMPL_V - 1) * SMPL_J;

    v8f acc = (v8f)(0.f);
#pragma unroll
    for (int k0 = 0; k0 < SMPL_J; k0 += 4) {
        int kr0 = k0 + 2 * khalf;
        v2f a = *(const v2f*)(wrow + kr0);          // weights, 8B aligned pairs
        v2f bb;
        bb.x = Ab[ccl + (unsigned)(kr0 * 12)];
        bb.y = Ab[ccl + (unsigned)(kr0 * 12 + 12)];
        acc = __builtin_amdgcn_wmma_f32_16x16x4_f32(
            false, a, false, bb, (short)0, acc, false, false);
    }

#pragma unroll
    for (int r = 0; r < 8; ++r) T[wave][r + 8 * khalf][c] = acc[r];
    __syncthreads();

    if (lane < 16) {
        int v = vtile + lane;
        if (v < SMPL_V) {
            const float* vp = vposed + (size_t)b * VPS + (size_t)v * 3;
            float x = vp[0], y = vp[1], z = vp[2];
            float* o = verts + (size_t)b * SMPL_N3 + (size_t)v * 3;
#pragma unroll
            for (int m = 0; m < 3; ++m) {
                float r0 = T[wave][lane][m * 4 + 0];
                float r1 = T[wave][lane][m * 4 + 1];
                float r2 = T[wave][lane][m * 4 + 2];
                float tr = T[wave][lane][m * 4 + 3];
                __builtin_nontemporal_store(
                    r0 * x + r1 * y + r2 * z + tr - rootp[b * 4 + m], o + m);
            }
        }
    }
}

// ---------------------------------------------------------------------------
extern "C" void kernel_launch(void* const* d_in, const int* in_sizes, int n_in,
                              void* d_out, int out_size, void* d_ws, size_t ws_size,
                              hipStream_t stream) {
    (void)n_in; (void)out_size; (void)ws_size;
    const float* poses   = (const float*)d_in[0];
    const float* betas   = (const float*)d_in[1];
    const float* kidb    = (const float*)d_in[2];
    const float* vt      = (const float*)d_in[3];
    const float* sd      = (const float*)d_in[4];
    const float* pd      = (const float*)d_in[5];
    const float* lw      = (const float*)d_in[6];
    const float* jr      = (const float*)d_in[7];
    const int*   parents = (const int*)d_in[8];

    int B = in_sizes[1] / SMPL_NB;   // 1024

    float* ws     = (float*)d_ws;
    float* JTJS   = ws;                                   // 864 (pad 1024)
    float* PF     = ws + 1024;                            // B*220
    float* Amat   = PF + (size_t)B * KPAD;                // B*288
    float* rootp  = Amat + (size_t)B * 288;               // B*4
    float* vposed = rootp + (size_t)B * 4;                // B*VPS (padded rows)

    float* verts     = (float*)d_out;
    float* joints    = verts + (size_t)B * SMPL_N3;
    float* poses_out = joints + (size_t)B * 72;

    jreg_kernel<<<72, 256, 0, stream>>>(jr, vt, sd, JTJS);
    pf_kernel<<<B, 256, 0, stream>>>(poses, betas, kidb, PF);
    joints_kernel<<<B, 32, 0, stream>>>(poses, betas, kidb, parents, JTJS,
                                        joints, poses_out, Amat, rootp);
    dim3 gC((SMPL_N3 + 511) / 512, B / 16);
    posegemm_kernel<<<gC, 128, 0, stream>>>(PF, pd, sd, vt, vposed);
    dim3 gD((SMPL_V + 127) / 128, B);
    lbs_kernel<<<gD, 256, 0, stream>>>(lw, Amat, vposed, rootp, verts);
}